// VolumeRenderer_73392401154454
// MI455X (gfx1250) — compile-verified
//
#include <hip/hip_runtime.h>

// Volume rendering (NeRF-style) forward pass for MI455X (gfx1250, wave32).
// Memory-bound: 256 MiB streamed reads -> ~11us at 23.3 TB/s. No GEMM
// structure (the feature/point operands vary per ray => no shared B matrix),
// so WMMA is inapplicable; the optimization targets are the data-movement
// path (coalesced b128 loads, non-temporal hints, global_prefetch_b8) and
// wave32 shuffle scan/reduce with minimal issue slots.

#define R_TOTAL   65536
#define N_SAMP    128
#define FAR_DELTA 1e10f

typedef float v4f __attribute__((ext_vector_type(4)));

__global__ __launch_bounds__(256) void volrend_fused_kernel(
    const float* __restrict__ depth_values,   // [R, N]
    const float* __restrict__ density,        // [R, N]
    const float* __restrict__ feature,        // [R, N, 3]
    const float* __restrict__ sample_points,  // [R, N, 3]
    float* __restrict__ out)                  // [R*3] feat ++ [R] depth
{
    const int lane            = threadIdx.x & 31;
    const int wave_in_block   = threadIdx.x >> 5;
    const int waves_per_block = blockDim.x >> 5;
    const int wave_id         = blockIdx.x * waves_per_block + wave_in_block;
    const int n_waves         = gridDim.x * waves_per_block;

    const int n0 = lane * 4;   // first of this lane's 4 samples

    for (int r = wave_id; r < R_TOTAL; r += n_waves) {
        // ---- prefetch next iteration's cache lines (global_prefetch_b8)
        const int rn = r + n_waves;
        if (rn < R_TOTAL) {
            const size_t bn = (size_t)rn * N_SAMP + (size_t)n0;
            __builtin_prefetch(depth_values  + bn,     0, 0);
            __builtin_prefetch(density       + bn,     0, 0);
            __builtin_prefetch(feature       + bn * 3, 0, 0);
            __builtin_prefetch(sample_points + bn * 3, 0, 0);
        }

        const size_t base = (size_t)r * N_SAMP + (size_t)n0;

        // ---- coalesced streaming loads (non-temporal: used exactly once)
        const v4f d4 = __builtin_nontemporal_load((const v4f*)(depth_values + base));
        const v4f s4 = __builtin_nontemporal_load((const v4f*)(density      + base));

        // depth of sample n0+4 lives in the next lane's d4.x
        const float nx = __shfl_down(d4.x, 1, 32);

        const float dl0 = d4.y - d4.x;
        const float dl1 = d4.z - d4.y;
        const float dl2 = d4.w - d4.z;
        const float dl3 = (lane == 31) ? FAR_DELTA : (nx - d4.w);

        // sig_delta = -sigma * delta
        const float g0 = -s4.x * dl0;
        const float g1 = -s4.y * dl1;
        const float g2 = -s4.z * dl2;
        const float g3 = -s4.w * dl3;

        // lane-local inclusive prefix over 4 elements
        const float p0 = g0;
        const float p1 = p0 + g1;
        const float p2 = p1 + g2;
        const float p3 = p2 + g3;

        // wave32 inclusive scan of lane sums (Hillis-Steele, 5 steps)
        float x = p3;
        #pragma unroll
        for (int off = 1; off < 32; off <<= 1) {
            const float y = __shfl_up(x, (unsigned)off, 32);
            if (lane >= off) x += y;
        }
        const float excl = x - p3;  // exclusive prefix entering this lane

        // trans_k = exp(prefix before elem k); w_k = exp(A_{k-1}) - exp(A_k)
        const float Em1 = __expf(excl);
        const float E0  = __expf(excl + p0);
        const float E1  = __expf(excl + p1);
        const float E2  = __expf(excl + p2);
        const float E3  = __expf(excl + p3);
        const float w0 = Em1 - E0;
        const float w1 = E0  - E1;
        const float w2 = E1  - E2;
        const float w3 = E2  - E3;

        // ---- feature: 12 contiguous floats per lane, 16B-aligned (48B chunks)
        const v4f* fp = (const v4f*)(feature + base * 3);
        const v4f fa = __builtin_nontemporal_load(fp + 0); // n0:c0 c1 c2 | n1:c0
        const v4f fb = __builtin_nontemporal_load(fp + 1); // n1:c1 c2 | n2:c0 c1
        const v4f fc = __builtin_nontemporal_load(fp + 2); // n2:c2 | n3:c0 c1 c2

        float ax = w0 * fa.x + w1 * fa.w + w2 * fb.z + w3 * fc.y;
        float ay = w0 * fa.y + w1 * fb.x + w2 * fb.w + w3 * fc.z;
        float az = w0 * fa.z + w1 * fb.y + w2 * fc.x + w3 * fc.w;

        // ---- depth: mean_n sigma * z ; z at stride 3 floats, offset +2
        const float* zp = sample_points + base * 3 + 2;
        const float z0 = __builtin_nontemporal_load(zp + 0);
        const float z1 = __builtin_nontemporal_load(zp + 3);
        const float z2 = __builtin_nontemporal_load(zp + 6);
        const float z3 = __builtin_nontemporal_load(zp + 9);
        float dz = s4.x * z0 + s4.y * z1 + s4.z * z2 + s4.w * z3;

        // ---- wave32 butterfly reductions (all lanes end with the totals)
        #pragma unroll
        for (int off = 16; off >= 1; off >>= 1) {
            ax += __shfl_xor(ax, off, 32);
            ay += __shfl_xor(ay, off, 32);
            az += __shfl_xor(az, off, 32);
            dz += __shfl_xor(dz, off, 32);
        }

        // ---- fan-out store: lanes 0..3 each write one result in a single
        // global_store_b32 issue (every lane holds the totals post-butterfly)
        if (lane < 4) {
            const float vout = (lane == 0) ? ax
                             : (lane == 1) ? ay
                             : (lane == 2) ? az
                             : dz * (1.0f / (float)N_SAMP);
            const int idx = (lane < 3) ? (3 * r + lane) : (3 * R_TOTAL + r);
            out[idx] = vout;
        }
    }
}

extern "C" void kernel_launch(void* const* d_in, const int* in_sizes, int n_in,
                              void* d_out, int out_size, void* d_ws, size_t ws_size,
                              hipStream_t stream) {
    (void)in_sizes; (void)n_in; (void)d_ws; (void)ws_size; (void)out_size;
    const float* depth_values  = (const float*)d_in[0];
    const float* density       = (const float*)d_in[1];
    const float* feature       = (const float*)d_in[2];
    const float* sample_points = (const float*)d_in[3];
    float* out = (float*)d_out;

    // 2048 blocks x 8 waves = 16384 waves; 4 rays per wave (grid-stride with
    // prefetch of the next ray's cache lines).
    dim3 grid(2048), block(256);
    volrend_fused_kernel<<<grid, block, 0, stream>>>(
        depth_values, density, feature, sample_points, out);
}